// OhemCrossEntropyLoss2d_9139690405890
// MI455X (gfx1250) — compile-verified
//
#include <hip/hip_runtime.h>
#include <math.h>

// Problem constants (from the reference)
#define NPIX (8 * 512 * 512)   // 2,097,152 pixels
#define CCH 19
#define HW (512 * 512)         // 262,144 (power of two: p>>18, p&(HW-1))
#define K_KEPT 100000u
#define OHEM_THRESH 0.7f
#define IGNORE_IDX 255
#define BLK 256
#define NBLK (NPIX / BLK)      // 8192 blocks

typedef __attribute__((ext_vector_type(2))) float v2f;
typedef __attribute__((ext_vector_type(8))) float v8f;

struct OhemState {
    unsigned prefix;     // radix-select bit prefix of kth-smallest value
    unsigned kRem;       // remaining rank within current prefix
    float threshold;     // final OHEM threshold
    float pad;
};

// Exact wave32 all-reduce (sum) on the matrix unit.
// WMMA #1: A = per-lane values (16x4 f32, a[1]=0), B = ones(4x16)
//   -> D[i][j] = v[i] + v[i+16]; each lane's 8 C/D VGPRs hold 8 distinct rows.
// Sum those 8 rows per lane, WMMA #2 with the same ones-B folds the two
// lane halves -> every lane of d2[0] holds the exact 32-lane sum.
// Requires EXEC == all ones (guaranteed at every call site: no divergence).
__device__ __forceinline__ float wave_allreduce_f32(float v) {
    v2f ones; ones[0] = 1.0f; ones[1] = 1.0f;
    v2f a;    a[0] = v;       a[1] = 0.0f;
    v8f z = {};
    v8f d = __builtin_amdgcn_wmma_f32_16x16x4_f32(false, a, false, ones,
                                                  (short)0, z, false, false);
    float s = d[0] + d[1] + d[2] + d[3] + d[4] + d[5] + d[6] + d[7];
    v2f a2; a2[0] = s; a2[1] = 0.0f;
    v8f d2 = __builtin_amdgcn_wmma_f32_16x16x4_f32(false, a2, false, ones,
                                                   (short)0, z, false, false);
    return d2[0];
}

// ---------------------------------------------------------------------------
// Kernel 0: zero the 4 radix histograms, init selection state.
// ---------------------------------------------------------------------------
__global__ void ohem_init(unsigned* __restrict__ hists, OhemState* __restrict__ st) {
    int t = threadIdx.x;
    if (t < 1024) hists[t] = 0u;
    if (t == 0) {
        st->prefix = 0u;
        st->kRem = K_KEPT;
        st->threshold = OHEM_THRESH;
    }
}

// ---------------------------------------------------------------------------
// Kernel 1: fused softmax / NLL / target-prob pass + radix histogram pass 0.
// One thread per pixel; 19 channel loads are wave-coalesced (channel stride
// HW, lanes contiguous in pixel index). Reads the 159 MB logits exactly once.
// ---------------------------------------------------------------------------
__global__ void ohem_softmax_pass(const float* __restrict__ logits,
                                  const int* __restrict__ target,
                                  float* __restrict__ mask_prob,
                                  float* __restrict__ nll,
                                  unsigned* __restrict__ hist0) {
    __shared__ unsigned lh[256];
    for (int i = threadIdx.x; i < 256; i += blockDim.x) lh[i] = 0u;
    __syncthreads();

    int p = blockIdx.x * blockDim.x + threadIdx.x;   // grid covers NPIX exactly
    int n = p >> 18;
    int rest = p & (HW - 1);
    size_t base = (size_t)n * (size_t)(CCH * HW) + (size_t)rest;

    float v[CCH];
#pragma unroll
    for (int c = 0; c < CCH; ++c) v[c] = logits[base + (size_t)c * HW];

    float m = v[0];
#pragma unroll
    for (int c = 1; c < CCH; ++c) m = fmaxf(m, v[c]);

    float se = 0.0f;
#pragma unroll
    for (int c = 0; c < CCH; ++c) se += expf(v[c] - m);

    int t = target[p];
    bool valid = (t != IGNORE_IDX);
    int tt = (valid && t >= 0 && t < CCH) ? t : 0;

    float lse = logf(se);
    float lp = v[tt] - m - lse;          // log p(target)
    float prob = expf(lp);

    float mp = valid ? prob : 1.0f;      // invalid pixels sort last (prob=1)
    mask_prob[p] = mp;
    nll[p] = valid ? -lp : -1.0f;        // NLL >= 0; -1 marks invalid

    // Radix pass 0 histogram: top byte of the (non-negative) float bits.
    atomicAdd(&lh[__float_as_uint(mp) >> 24], 1u);
    __syncthreads();
    for (int i = threadIdx.x; i < 256; i += blockDim.x)
        if (lh[i]) atomicAdd(&hist0[i], lh[i]);
}

// ---------------------------------------------------------------------------
// Kernel 2: radix counting pass (shift = 16, 8, 0). Streams mask_prob (8 MB).
// ---------------------------------------------------------------------------
__global__ void ohem_radix_count(const float* __restrict__ mask_prob,
                                 const OhemState* __restrict__ st,
                                 unsigned* __restrict__ hist,
                                 int shift) {
    __shared__ unsigned lh[256];
    for (int i = threadIdx.x; i < 256; i += blockDim.x) lh[i] = 0u;
    __syncthreads();

    unsigned prefix = st->prefix;
    int p = blockIdx.x * blockDim.x + threadIdx.x;
    unsigned u = __float_as_uint(mask_prob[p]);
    if ((u >> (shift + 8)) == prefix)
        atomicAdd(&lh[(u >> shift) & 0xFFu], 1u);

    __syncthreads();
    for (int i = threadIdx.x; i < 256; i += blockDim.x)
        if (lh[i]) atomicAdd(&hist[i], lh[i]);
}

// ---------------------------------------------------------------------------
// Kernel 3: pick the byte containing the kRem-th remaining value.
// ---------------------------------------------------------------------------
__global__ void ohem_radix_select(const unsigned* __restrict__ hist,
                                  OhemState* __restrict__ st, int last) {
    unsigned kRem = st->kRem;
    unsigned cum = 0;
    int b = 0;
    for (; b < 256; ++b) {
        unsigned h = hist[b];
        if (cum + h >= kRem) break;
        cum += h;
    }
    if (b > 255) b = 255;
    st->prefix = (st->prefix << 8) | (unsigned)b;
    st->kRem = kRem - cum;
    if (last) {
        float kth = __uint_as_float(st->prefix);   // exact kth-smallest value
        st->threshold = fmaxf(kth, OHEM_THRESH);   // where(kth>0.7, kth, 0.7)
    }
}

// ---------------------------------------------------------------------------
// Kernel 4: keep-mask + per-block partial sums (deterministic: WMMA wave
// reduce -> LDS -> thread 0). Count via ballot/popc (exact, deterministic).
// ---------------------------------------------------------------------------
__global__ void ohem_keep_sum(const float* __restrict__ mask_prob,
                              const float* __restrict__ nll,
                              const OhemState* __restrict__ st,
                              float* __restrict__ psum,
                              float* __restrict__ pcnt) {
    __shared__ float wsum[BLK / 32];
    __shared__ unsigned wcnt[BLK / 32];

    float thr = st->threshold;
    int p = blockIdx.x * blockDim.x + threadIdx.x;
    float mp = mask_prob[p];
    float nl = nll[p];
    bool keep = (nl >= 0.0f) && (mp <= thr);   // valid & kept
    float val = keep ? nl : 0.0f;

    float s = wave_allreduce_f32(val);               // v_wmma_f32_16x16x4_f32
    unsigned long long bal = __ballot(keep);         // wave32 -> low 32 bits
    int lane = threadIdx.x & 31;
    int wid = threadIdx.x >> 5;
    if (lane == 0) { wsum[wid] = s; wcnt[wid] = (unsigned)__popcll(bal); }
    __syncthreads();
    if (threadIdx.x == 0) {
        float ts = 0.0f; unsigned tc = 0u;
        for (int i = 0; i < BLK / 32; ++i) { ts += wsum[i]; tc += wcnt[i]; }
        psum[blockIdx.x] = ts;
        pcnt[blockIdx.x] = (float)tc;
    }
}

// ---------------------------------------------------------------------------
// Kernel 5: fixed-order final reduction of 8192 partials -> loss scalar.
// ---------------------------------------------------------------------------
__global__ void ohem_final(const float* __restrict__ psum,
                           const float* __restrict__ pcnt,
                           float* __restrict__ out) {
    __shared__ float ssum[8];
    __shared__ float scnt[8];
    float s = 0.0f, c = 0.0f;
    for (int i = threadIdx.x; i < NBLK; i += 256) {  // 32 iters, no divergence
        s += psum[i];
        c += pcnt[i];
    }
    float rs = wave_allreduce_f32(s);
    float rc = wave_allreduce_f32(c);
    int lane = threadIdx.x & 31;
    int wid = threadIdx.x >> 5;
    if (lane == 0) { ssum[wid] = rs; scnt[wid] = rc; }
    __syncthreads();
    if (threadIdx.x == 0) {
        float ts = 0.0f, tc = 0.0f;
        for (int i = 0; i < 8; ++i) { ts += ssum[i]; tc += scnt[i]; }
        out[0] = ts / fmaxf(tc, 1.0f);
    }
}

// ---------------------------------------------------------------------------
// Launch: init -> fused softmax(+hist0) -> 4x radix select (3 count passes)
// -> keep/sum -> final. All on `stream`; workspace layout below (~16.1 MB).
// ---------------------------------------------------------------------------
extern "C" void kernel_launch(void* const* d_in, const int* in_sizes, int n_in,
                              void* d_out, int out_size, void* d_ws, size_t ws_size,
                              hipStream_t stream) {
    (void)in_sizes; (void)n_in; (void)out_size; (void)ws_size;

    const float* logits = (const float*)d_in[0];
    const int*   target = (const int*)d_in[1];
    float* out = (float*)d_out;

    char* ws = (char*)d_ws;
    float*     mask_prob = (float*)ws;                              // 8 MB
    float*     nll       = (float*)(ws + (size_t)NPIX * 4);         // 8 MB
    unsigned*  hists     = (unsigned*)(ws + (size_t)NPIX * 8);      // 4 KB (4x256)
    OhemState* st        = (OhemState*)(ws + (size_t)NPIX * 8 + 4096);
    float*     psum      = (float*)(ws + (size_t)NPIX * 8 + 4096 + 64);  // 32 KB
    float*     pcnt      = psum + NBLK;                                  // 32 KB

    ohem_init<<<1, 1024, 0, stream>>>(hists, st);
    ohem_softmax_pass<<<NBLK, BLK, 0, stream>>>(logits, target, mask_prob, nll,
                                                hists + 0);
    ohem_radix_select<<<1, 1, 0, stream>>>(hists + 0, st, 0);
    ohem_radix_count<<<NBLK, BLK, 0, stream>>>(mask_prob, st, hists + 256, 16);
    ohem_radix_select<<<1, 1, 0, stream>>>(hists + 256, st, 0);
    ohem_radix_count<<<NBLK, BLK, 0, stream>>>(mask_prob, st, hists + 512, 8);
    ohem_radix_select<<<1, 1, 0, stream>>>(hists + 512, st, 0);
    ohem_radix_count<<<NBLK, BLK, 0, stream>>>(mask_prob, st, hists + 768, 0);
    ohem_radix_select<<<1, 1, 0, stream>>>(hists + 768, st, 1);
    ohem_keep_sum<<<NBLK, BLK, 0, stream>>>(mask_prob, nll, st, psum, pcnt);
    ohem_final<<<1, 256, 0, stream>>>(psum, pcnt, out);
}